// Property_46248207843985
// MI455X (gfx1250) — compile-verified
//
#include <hip/hip_runtime.h>
#include <math.h>

typedef __attribute__((ext_vector_type(2))) float v2f;
typedef __attribute__((ext_vector_type(8))) float v8f;

#define NCONF   8
#define NATOMS  10
#define NSTATES 4
#define NBASIS  24
#define HIDDEN  48
#define NPAIR   90      // 10*9 ordered pairs per conf
#define NDIR    30      // natoms*3 perturbation directions per conf
#define CUTOFF  6.0f
#define ALPHA   1.5f
#define PI_F    3.14159265358979323846f

// ---------------------------------------------------------------------------
// Kernel 1: one block per (conf, direction). Computes 2nd-order Taylor jets
// (value, d/dt, d2/dt2) of tmppsi[s] along coordinate x[m,l] of conf c.
// Two wave32 waves; MLP layers done with V_WMMA_F32_16X16X4_F32.
// Jet rows in the M dimension: rows 0..9 value, 10..19 first, 20..29 second,
// rows 30..31 zero padding -> exactly 2 M-tiles of 16.
// ---------------------------------------------------------------------------
__global__ __launch_bounds__(64) void wave_jet_kernel(
    const float* __restrict__ cart,    // [8][10][3]
    const int*   __restrict__ species, // [8][10]
    const float* __restrict__ W1,      // [24][48]
    const float* __restrict__ B1,      // [4][48]
    const float* __restrict__ W2,      // [48][4]
    const float* __restrict__ Emb,     // [4][4]
    float*       __restrict__ ws)      // [8*30][4][3]
{
  const int blk = blockIdx.x;
  const int c   = blk / NDIR;
  const int dir = blk % NDIR;
  const int m   = dir / 3;      // perturbed atom
  const int l   = dir % 3;      // perturbed coordinate
  const int t   = threadIdx.x;

  __shared__ float s_cart[NATOMS * 3];
  __shared__ int   s_spec[NATOMS];
  __shared__ float s_rj[NPAIR * 3];        // r jets  (rv, rd, re)
  __shared__ float s_Rj[NPAIR * 3];        // radial jets (Rv, Rd, Re)
  __shared__ float s_dens[32 * NBASIS];    // jet rows x 24  (A of layer 1)
  __shared__ float s_pre[32 * 50];         // layer-1 pre-activation (padded)
  __shared__ float s_h[32 * 50];           // tanh-jet rows   (A of layer 2)
  __shared__ float s_ip[32 * 16];          // initpsi jets (N padded to 16)
  __shared__ float s_tmp[NSTATES * 3];     // tmppsi jets

  if (t == 0) { __builtin_prefetch(W1, 0, 0); __builtin_prefetch(W2, 0, 0); }

  if (t < NATOMS * 3) s_cart[t] = cart[c * NATOMS * 3 + t];
  if (t < NATOMS)     s_spec[t] = species[c * NATOMS + t];
  if (t < NSTATES * 3) s_tmp[t] = 0.0f;
  // zero the padding rows (30,31) of the two WMMA A operands
  if (t < 48) s_dens[30 * NBASIS + t] = 0.0f;
  for (int z = t; z < 100; z += 64) s_h[30 * 50 + z] = 0.0f;
  __syncthreads();

  // ---- phase A: pair distance jets + radial jets ----
  const float kc = PI_F / CUTOFF;
  for (int p = t; p < NPAIR; p += 64) {
    int i  = p / 9;
    int jj = p % 9;
    int j  = jj + (jj >= i ? 1 : 0);
    float dx = s_cart[j * 3 + 0] - s_cart[i * 3 + 0];
    float dy = s_cart[j * 3 + 1] - s_cart[i * 3 + 1];
    float dz = s_cart[j * 3 + 2] - s_cart[i * 3 + 2];
    float rv = sqrtf(dx * dx + dy * dy + dz * dz);
    float sd = (j == m ? 1.0f : 0.0f) - (i == m ? 1.0f : 0.0f);
    float dl = (l == 0) ? dx : ((l == 1) ? dy : dz);
    float rd = sd * dl / rv;                 // dr/dt
    float re = (sd * sd - rd * rd) / rv;     // d2r/dt2
    s_rj[p * 3 + 0] = rv;
    s_rj[p * 3 + 1] = rd;
    s_rj[p * 3 + 2] = re;
    // radial = sin(k r)^3 jets
    float sn = sinf(kc * rv), cs = cosf(kc * rv);
    float Rv  = sn * sn * sn;
    float Rp  = 3.0f * sn * sn * cs * kc;
    float Rpp = 3.0f * kc * kc * sn * (2.0f * cs * cs - sn * sn);
    s_Rj[p * 3 + 0] = Rv;
    s_Rj[p * 3 + 1] = Rp * rd;
    s_Rj[p * 3 + 2] = Rpp * rd * rd + Rp * re;
  }
  __syncthreads();

  // ---- phase B: density jets (segment-sum of RBF*fcut over 9 neighbors) ----
  for (int e = t; e < NATOMS * NBASIS; e += 64) {
    int i = e / NBASIS, b = e % NBASIS;
    float cb = CUTOFF * (float)b / (float)(NBASIS - 1);
    float sv = 0.0f, sdv = 0.0f, sev = 0.0f;
    for (int jj = 0; jj < 9; ++jj) {
      int p = i * 9 + jj;
      float rv = s_rj[p * 3 + 0], rd = s_rj[p * 3 + 1], re = s_rj[p * 3 + 2];
      float del = rv - cb;
      float g   = expf(-ALPHA * del * del);
      float gp  = -2.0f * ALPHA * del * g;
      float gpp = (4.0f * ALPHA * ALPHA * del * del - 2.0f * ALPHA) * g;
      float f, fp, fpp;
      if (rv < CUTOFF) {
        f   = 0.5f * (cosf(kc * rv) + 1.0f);
        fp  = -0.5f * kc * sinf(kc * rv);
        fpp = -0.5f * kc * kc * cosf(kc * rv);
      } else { f = 0.0f; fp = 0.0f; fpp = 0.0f; }   // min(r,C) clamps grads
      float phi   = g * f;
      float phip  = gp * f + g * fp;
      float phipp = gpp * f + 2.0f * gp * fp + g * fpp;
      sv  += phi;
      sdv += phip * rd;
      sev += phipp * rd * rd + phip * re;
    }
    s_dens[(0  + i) * NBASIS + b] = sv;
    s_dens[(10 + i) * NBASIS + b] = sdv;
    s_dens[(20 + i) * NBASIS + b] = sev;
  }
  __syncthreads();

  // ---- phase C: layer 1, [32 x 24] @ [24 x 48] via WMMA F32 16x16x4 ----
  const int wave  = t >> 5;
  const int lane  = t & 31;
  const int mrow  = (wave << 4) + (lane & 15); // A row for this lane
  const int ksel  = (lane >> 4) << 1;          // 0 for lanes 0-15, 2 for 16-31
  const int col   = lane & 15;

  v8f acc0 = {}, acc1 = {}, acc2 = {};
  for (int k0 = 0; k0 < NBASIS; k0 += 4) {
    v2f a;
    a.x = s_dens[mrow * NBASIS + k0 + ksel];
    a.y = s_dens[mrow * NBASIS + k0 + ksel + 1];
    v2f b0, b1, b2;
    b0.x = W1[(k0 + ksel    ) * HIDDEN +  0 + col];
    b0.y = W1[(k0 + ksel + 1) * HIDDEN +  0 + col];
    b1.x = W1[(k0 + ksel    ) * HIDDEN + 16 + col];
    b1.y = W1[(k0 + ksel + 1) * HIDDEN + 16 + col];
    b2.x = W1[(k0 + ksel    ) * HIDDEN + 32 + col];
    b2.y = W1[(k0 + ksel + 1) * HIDDEN + 32 + col];
    acc0 = __builtin_amdgcn_wmma_f32_16x16x4_f32(false, a, false, b0, (short)0, acc0, false, false);
    acc1 = __builtin_amdgcn_wmma_f32_16x16x4_f32(false, a, false, b1, (short)0, acc1, false, false);
    acc2 = __builtin_amdgcn_wmma_f32_16x16x4_f32(false, a, false, b2, (short)0, acc2, false, false);
  }
  // write C fragments: VGPR v holds M = v (lanes<16) / v+8 (lanes>=16)
  {
    int rbase = (wave << 4) + ((lane >> 4) << 3);
#pragma unroll
    for (int v = 0; v < 8; ++v) {
      int row = rbase + v;
      s_pre[row * 50 +  0 + col] = acc0[v];
      s_pre[row * 50 + 16 + col] = acc1[v];
      s_pre[row * 50 + 32 + col] = acc2[v];
    }
  }
  __syncthreads();

  // ---- phase D: tanh jet (bias only on value rows) ----
  for (int e = t; e < NATOMS * HIDDEN; e += 64) {
    int i = e / HIDDEN, hc = e % HIDDEN;
    float pv = s_pre[(0  + i) * 50 + hc] + B1[s_spec[i] * HIDDEN + hc];
    float pd = s_pre[(10 + i) * 50 + hc];
    float pe = s_pre[(20 + i) * 50 + hc];
    float hv = tanhf(pv);
    float om = 1.0f - hv * hv;
    s_h[(0  + i) * 50 + hc] = hv;
    s_h[(10 + i) * 50 + hc] = om * pd;
    s_h[(20 + i) * 50 + hc] = om * pe - 2.0f * hv * om * pd * pd;
  }
  __syncthreads();

  // ---- phase E: layer 2, [32 x 48] @ [48 x 4(->16)] via WMMA ----
  v8f accp = {};
  for (int k0 = 0; k0 < HIDDEN; k0 += 4) {
    v2f a;
    a.x = s_h[mrow * 50 + k0 + ksel];
    a.y = s_h[mrow * 50 + k0 + ksel + 1];
    v2f b;
    b.x = (col < NSTATES) ? W2[(k0 + ksel    ) * NSTATES + col] : 0.0f;
    b.y = (col < NSTATES) ? W2[(k0 + ksel + 1) * NSTATES + col] : 0.0f;
    accp = __builtin_amdgcn_wmma_f32_16x16x4_f32(false, a, false, b, (short)0, accp, false, false);
  }
  {
    int rbase = (wave << 4) + ((lane >> 4) << 3);
#pragma unroll
    for (int v = 0; v < 8; ++v)
      s_ip[(rbase + v) * 16 + col] = accp[v];
  }
  __syncthreads();

  // ---- phase F: radial-weighted gather/scatter jet -> tmppsi jets ----
  for (int p = t; p < NPAIR; p += 64) {
    int i  = p / 9;
    int jj = p % 9;
    int j  = jj + (jj >= i ? 1 : 0);
    float Rv = s_Rj[p * 3 + 0], Rd = s_Rj[p * 3 + 1], Re = s_Rj[p * 3 + 2];
#pragma unroll
    for (int s = 0; s < NSTATES; ++s) {
      float iv = s_ip[(0  + j) * 16 + s] + Emb[s_spec[j] * NSTATES + s];
      float id = s_ip[(10 + j) * 16 + s];
      float ie = s_ip[(20 + j) * 16 + s];
      atomicAdd(&s_tmp[s * 3 + 0], iv * Rv);
      atomicAdd(&s_tmp[s * 3 + 1], id * Rv + iv * Rd);
      atomicAdd(&s_tmp[s * 3 + 2], ie * Rv + 2.0f * id * Rd + iv * Re);
    }
  }
  __syncthreads();

  if (t < NSTATES * 3) ws[blk * 12 + t] = s_tmp[t];
}

// ---------------------------------------------------------------------------
// Kernel 2: reduce 240 direction-jets into the final scalar loss.
// psi[0] = T0^2 => Laplacian(psi0) = 2*grad^2 + 2*T0*Lap(T0); states 1..3
// are linear. kin = -0.5 * sum_{m,l} massrev * d2psi; vib = (kin+pot*psi)/psi.
// ---------------------------------------------------------------------------
__global__ __launch_bounds__(64) void vibene_reduce_kernel(
    const float* __restrict__ ws,       // [8*30][4][3]
    const float* __restrict__ pot,      // [8]
    const float* __restrict__ massrev,  // [8][10]
    const float* __restrict__ elevel,   // [4]
    float*       __restrict__ out)
{
  __shared__ float s_acc;
  int t = threadIdx.x;
  if (t == 0) s_acc = 0.0f;
  __syncthreads();
  if (t < NCONF * NSTATES) {
    int cidx = t >> 2, s = t & 3;
    float T = ws[(cidx * NDIR + 0) * 12 + s * 3 + 0];  // value: same all dirs
    float Hm = 0.0f, G2 = 0.0f;
    for (int dir = 0; dir < NDIR; ++dir) {
      float mr = massrev[cidx * NATOMS + dir / 3];
      const float* q = &ws[(cidx * NDIR + dir) * 12 + s * 3];
      float dd = q[1], ee = q[2];
      Hm += mr * ee;
      G2 += mr * dd * dd;
    }
    float psi, L;
    if (s == 0) { psi = T * T; L = 2.0f * G2 + 2.0f * T * Hm; }
    else        { psi = T;     L = Hm; }
    float kin    = -0.5f * L;                  // FACTOR_KIN
    float potene = pot[cidx] - pot[NCONF - 1];
    float vib    = (kin + psi * potene) / psi;
    float d      = vib - elevel[s];
    atomicAdd(&s_acc, d * d);
  }
  __syncthreads();
  if (t == 0) out[0] = s_acc;
}

extern "C" void kernel_launch(void* const* d_in, const int* in_sizes, int n_in,
                              void* d_out, int out_size, void* d_ws, size_t ws_size,
                              hipStream_t stream) {
  (void)in_sizes; (void)n_in; (void)out_size; (void)ws_size;
  // setup_inputs order:
  // 0 eigen_weight (unused by reference), 1 pot, 2 cart, 3 numatoms (unused),
  // 4 species, 5 massrev, 6 atom_index (structure known), 7 shifts (zero),
  // 8 W1, 9 B1, 10 W2, 11 Emb, 12 elevel
  const float* pot     = (const float*)d_in[1];
  const float* cart    = (const float*)d_in[2];
  const int*   species = (const int*)  d_in[4];
  const float* massrev = (const float*)d_in[5];
  const float* W1      = (const float*)d_in[8];
  const float* B1      = (const float*)d_in[9];
  const float* W2      = (const float*)d_in[10];
  const float* Emb     = (const float*)d_in[11];
  const float* elevel  = (const float*)d_in[12];
  float* ws  = (float*)d_ws;           // 240 * 12 floats = 11.25 KB
  float* out = (float*)d_out;

  wave_jet_kernel<<<NCONF * NDIR, 64, 0, stream>>>(cart, species, W1, B1, W2, Emb, ws);
  vibene_reduce_kernel<<<1, 64, 0, stream>>>(ws, pot, massrev, elevel, out);
}